// PairwisePotential_25812753449050
// MI455X (gfx1250) — compile-verified
//
#include <hip/hip_runtime.h>
#include <stdint.h>

// Problem constants (match reference)
#define HDIM   512
#define BDIM   4
#define CDIM   8
#define TILE   32
#define HALO   4                    // max dilation
#define LDSW   (TILE + 2*HALO)      // 40
#define NTILE  (HDIM / TILE)        // 16
#define SLICES (BDIM * CDIM)        // 32
#define NBLK   (NTILE * NTILE * SLICES) // 8192

// ---- CDNA5 async global->LDS copy (gfx1250), guarded for portability ----
#if defined(__has_builtin)
#  if __has_builtin(__builtin_amdgcn_global_load_async_to_lds_b32) && \
      __has_builtin(__builtin_amdgcn_s_wait_asynccnt)
#    define HAVE_ASYNC_LDS 1
#  endif
#endif
#ifndef HAVE_ASYNC_LDS
#  define HAVE_ASYNC_LDS 0
#endif

#if HAVE_ASYNC_LDS
typedef __attribute__((address_space(1))) int* glb_i32_ptr;
typedef __attribute__((address_space(3))) int* lds_i32_ptr;
#define ASYNC_CP_B32(dst_lds, src_glb)                                   \
  __builtin_amdgcn_global_load_async_to_lds_b32(                         \
      (glb_i32_ptr)(src_glb), (lds_i32_ptr)(dst_lds), 0, 0)
#endif

// exp(-0.5*diff^2 - 0.5*dist) = 2^(-(s*diff)^2 + lw)
//   with s  = sqrt(0.5*log2(e)) = 0.84932180
//   and  lw = -0.5*log2(e)*dist
#define PRESCALE 0.84932180f

// Inputs in LDS are pre-scaled by PRESCALE. Per tap: sub, fma(neg), exp2, add.
static __device__ __forceinline__ float pot_taps(const float* __restrict__ s,
                                                 int cidx, float c) {
  // lw[dx][dy] = -0.72134752 * sqrt(dx^2+dy^2), dx,dy in 0..2
  const float Lw[3][3] = {
      { 0.0f,         -0.72134752f, -1.44269504f},
      {-0.72134752f,  -1.02013940f, -1.61298407f},
      {-1.44269504f,  -1.61298407f, -2.04027880f}};
  float acc = 1.47920607f;  // center tap (diff=0, dist=sqrt(2)) x 3 dilations
#pragma unroll
  for (int di = 0; di < 3; ++di) {
    const int d = (di == 0) ? 1 : (di == 1) ? 2 : 4;
#pragma unroll
    for (int i = 0; i < 3; ++i) {
#pragma unroll
      for (int j = 0; j < 3; ++j) {
        if (i == 1 && j == 1) continue;  // center handled as constant
        const float ds_ = c - s[cidx + (i - 1) * d * LDSW + (j - 1) * d];
        // v_fma_f32 with src0 neg-modifier; single literal operand
        const float e = __builtin_amdgcn_exp2f(fmaf(-ds_, ds_, Lw[i][j]));
        acc += e;
      }
    }
  }
  return acc;
}

__global__ __launch_bounds__(256) void pot_mse_kernel(
    const float* __restrict__ x, const float* __restrict__ lab,
    float* __restrict__ parts) {
  __shared__ float sx[LDSW * LDSW];
  __shared__ float sl[LDSW * LDSW];
  __shared__ float red[256];

  const int tid = threadIdx.y * 32 + threadIdx.x;
  const size_t base = (size_t)blockIdx.z * HDIM * HDIM;
  const int gy0 = blockIdx.y * TILE - HALO;
  const int gx0 = blockIdx.x * TILE - HALO;
  const float* xs = x + base;
  const float* ls = lab + base;

  // Stage both halo tiles into LDS (zero-fill = jnp.pad semantics).
  for (int idx = tid; idx < LDSW * LDSW; idx += 256) {
    const int ly = idx / LDSW;
    const int lx = idx - ly * LDSW;
    const int gy = gy0 + ly;
    const int gx = gx0 + lx;
    if ((unsigned)gy < HDIM && (unsigned)gx < HDIM) {
      const size_t off = (size_t)gy * HDIM + gx;
#if HAVE_ASYNC_LDS
      ASYNC_CP_B32(&sx[idx], xs + off);
      ASYNC_CP_B32(&sl[idx], ls + off);
#else
      sx[idx] = xs[off];
      sl[idx] = ls[off];
#endif
    } else {
      sx[idx] = 0.0f;
      sl[idx] = 0.0f;
    }
  }
#if HAVE_ASYNC_LDS
  __builtin_amdgcn_s_wait_asynccnt(0);  // wave-local async completion
#endif
  __syncthreads();  // all waves' tiles staged

  // Pre-scale tiles so the per-tap exponent is a single fma (amortized cost).
  for (int idx = tid; idx < LDSW * LDSW; idx += 256) {
    sx[idx] *= PRESCALE;
    sl[idx] *= PRESCALE;
  }
  __syncthreads();

  // Each thread computes 4 pixels (32x8 block covers 32x32 tile in 4 row-steps)
  float psum = 0.0f;
#pragma unroll
  for (int r = 0; r < 4; ++r) {
    const int row = threadIdx.y + 8 * r;
    const int cidx = (HALO + row) * LDSW + HALO + threadIdx.x;
    const float gx_ = pot_taps(sx, cidx, sx[cidx]);
    const float gl_ = pot_taps(sl, cidx, sl[cidx]);
    const float d = gx_ - gl_;
    psum = fmaf(d, d, psum);
  }

  // Deterministic block reduction
  red[tid] = psum;
  __syncthreads();
#pragma unroll
  for (int s = 128; s > 0; s >>= 1) {
    if (tid < s) red[tid] += red[tid + s];
    __syncthreads();
  }
  if (tid == 0) {
    parts[(blockIdx.z * gridDim.y + blockIdx.y) * gridDim.x + blockIdx.x] =
        red[0];
  }
}

__global__ __launch_bounds__(256) void final_reduce_kernel(
    const float* __restrict__ parts, float* __restrict__ out, int n,
    float scale) {
  __shared__ float red[256];
  float s = 0.0f;
  for (int i = threadIdx.x; i < n; i += 256) s += parts[i];  // fixed order
  red[threadIdx.x] = s;
  __syncthreads();
#pragma unroll
  for (int k = 128; k > 0; k >>= 1) {
    if (threadIdx.x < k) red[threadIdx.x] += red[threadIdx.x + k];
    __syncthreads();
  }
  if (threadIdx.x == 0) out[0] = red[0] * scale;
}

extern "C" void kernel_launch(void* const* d_in, const int* in_sizes, int n_in,
                              void* d_out, int out_size, void* d_ws,
                              size_t ws_size, hipStream_t stream) {
  (void)in_sizes; (void)n_in; (void)out_size; (void)ws_size;
  const float* x = (const float*)d_in[0];
  // d_in[1] (label_single) is unused by the reference output.
  const float* label = (const float*)d_in[2];
  float* parts = (float*)d_ws;  // needs NBLK*4 = 32 KB scratch

  dim3 grid(NTILE, NTILE, SLICES);
  dim3 block(32, 8);
  pot_mse_kernel<<<grid, block, 0, stream>>>(x, label, parts);

  const float scale = 1.0f / (float)((size_t)BDIM * CDIM * HDIM * HDIM);
  final_reduce_kernel<<<1, 256, 0, stream>>>(parts, (float*)d_out, NBLK, scale);
}